// NormalEstimator_82970178224388
// MI455X (gfx1250) — compile-verified
//
#include <hip/hip_runtime.h>
#include <hip/hip_bf16.h>
#include <math.h>

typedef __attribute__((ext_vector_type(2))) float v2f;
typedef __attribute__((ext_vector_type(8))) float v8f;

#define NPTS   100000
#define PBS    500
#define KNN    100
#define NKP    15
#define ROWS   (PBS*KNN)      /* 50000 */
#define KPAD   48             /* 45 padded to multiple of 4 */
#define C1     128
#define C2     1024
#define EPSBN  1e-5f

/* ------------------------------------------------------------------ */
/* kNN: exact top-100 smallest L-inf distances per query.             */
/* Histogram select on top-13 float bits, tie-break by index via      */
/* 64-bit key (dist_bits<<32 | idx). Neighbor ORDER does not matter   */
/* downstream (all K-reductions are symmetric).                       */
/* ------------------------------------------------------------------ */
#define HBINS   8192
#define MAXCAND 2048

__global__ void ne_knn_kernel(const float* __restrict__ pts,
                              const int*   __restrict__ index,
                              int*         __restrict__ nbr)
{
  __shared__ unsigned hist[HBINS];
  __shared__ unsigned long long cand[MAXCAND];
  __shared__ int s_pivot, s_below, s_ncand, s_nacc;
  __shared__ float s_q[3];

  const int q   = blockIdx.x;
  const int tid = threadIdx.x;
  const int bs  = blockDim.x;

  if (tid == 0) {
    int si = index[q];
    s_q[0] = pts[si*3+0]; s_q[1] = pts[si*3+1]; s_q[2] = pts[si*3+2];
    s_ncand = 0; s_nacc = 0;
  }
  for (int i = tid; i < HBINS; i += bs) hist[i] = 0;
  __syncthreads();

  const float qx = s_q[0], qy = s_q[1], qz = s_q[2];

  for (int i = tid; i < NPTS; i += bs) {
    float dx = fabsf(pts[i*3+0]-qx);
    float dy = fabsf(pts[i*3+1]-qy);
    float dz = fabsf(pts[i*3+2]-qz);
    float d  = fmaxf(fmaxf(dx,dy),dz);
    atomicAdd(&hist[__float_as_uint(d) >> 19], 1u);
  }
  __syncthreads();

  if (tid == 0) {
    unsigned cum = 0; int pivot = HBINS-1; unsigned below = 0;
    for (int bin = 0; bin < HBINS; ++bin) {
      unsigned c = hist[bin];
      if (cum + c >= (unsigned)KNN) { pivot = bin; below = cum; break; }
      cum += c;
    }
    s_pivot = pivot; s_below = (int)below;
  }
  __syncthreads();

  const int pivot = s_pivot;
  for (int i = tid; i < NPTS; i += bs) {
    float dx = fabsf(pts[i*3+0]-qx);
    float dy = fabsf(pts[i*3+1]-qy);
    float dz = fabsf(pts[i*3+2]-qz);
    float d  = fmaxf(fmaxf(dx,dy),dz);
    unsigned bits = __float_as_uint(d);
    int bin = (int)(bits >> 19);
    if (bin < pivot) {
      int pos = atomicAdd(&s_nacc, 1);
      nbr[q*KNN + pos] = i;
    } else if (bin == pivot) {
      int cpos = atomicAdd(&s_ncand, 1);
      if (cpos < MAXCAND)
        cand[cpos] = ((unsigned long long)bits << 32) | (unsigned)i;
    }
  }
  __syncthreads();

  int nc = s_ncand; if (nc > MAXCAND) nc = MAXCAND;
  const int m  = KNN - s_below;
  for (int j = tid; j < nc; j += bs) {
    unsigned long long key = cand[j];
    int rank = 0;
    for (int t = 0; t < nc; ++t) rank += (cand[t] < key) ? 1 : 0;
    if (rank < m)
      nbr[q*KNN + s_below + rank] = (int)(unsigned)(key & 0xffffffffu);
  }
}

/* ------------------------------------------------------------------ */
/* Init stats buffers (done every launch: ws is not re-poisoned).     */
/* ------------------------------------------------------------------ */
__global__ void ne_init_kernel(float* __restrict__ sum, float* __restrict__ sumsq,
                               unsigned* __restrict__ maxb, unsigned* __restrict__ minb)
{
  int idx = blockIdx.x*blockDim.x + threadIdx.x;
  if (idx < PBS*C2) { maxb[idx] = 0u; minb[idx] = 0x7f800000u; }
  if (idx < C2)     { sum[idx] = 0.f; sumsq[idx] = 0.f; }
}

/* Transposed+padded KP weights: wkT[o*48 + kk] = kpw[kk*128 + o], kk<45 */
__global__ void ne_padwk_kernel(const float* __restrict__ kpw, float* __restrict__ wkT)
{
  int idx = blockIdx.x*blockDim.x + threadIdx.x;       /* 128*48 */
  if (idx >= C1*KPAD) return;
  int o = idx / KPAD, kk = idx % KPAD;
  wkT[idx] = (kk < 45) ? kpw[kk*C1 + o] : 0.f;
}

/* ------------------------------------------------------------------ */
/* rel + gaussian kernel weights -> wf [50000 x 48] (45 real + pad)   */
/* ------------------------------------------------------------------ */
__global__ void ne_wf_kernel(const float* __restrict__ pts,
                             const int*   __restrict__ index,
                             const int*   __restrict__ nbr,
                             const float* __restrict__ kpp,
                             const float* __restrict__ sigma_p,
                             float*       __restrict__ wf)
{
  int r = blockIdx.x*blockDim.x + threadIdx.x;
  if (r >= ROWS) return;
  int b  = r / KNN;
  int i  = nbr[r];
  int si = index[b];
  float rx = pts[i*3+0] - pts[si*3+0];
  float ry = pts[i*3+1] - pts[si*3+1];
  float rz = pts[i*3+2] - pts[si*3+2];
  float sg = sigma_p[0];
  float c  = 0.5f / (sg*sg);
  float* row = wf + (size_t)r * KPAD;
  #pragma unroll
  for (int kp = 0; kp < NKP; ++kp) {
    float dx = rx - kpp[kp*3+0];
    float dy = ry - kpp[kp*3+1];
    float dz = rz - kpp[kp*3+2];
    float w  = expf(-(dx*dx + dy*dy + dz*dz) * c);
    row[kp*3+0] = rx*w; row[kp*3+1] = ry*w; row[kp*3+2] = rz*w;
  }
  row[45] = 0.f; row[46] = 0.f; row[47] = 0.f;
}

/* ------------------------------------------------------------------ */
/* Generic f32 WMMA GEMM: C[M,Nout] = act(A[M,K] * W[Nout,K]^T + b)   */
/* One wave per 16x64 tile, 4 independent accumulator chains sharing  */
/* one A-fragment, with REGISTER DOUBLE-BUFFERING: next K-step's 5    */
/* fragments are loaded before the current step's 4 WMMAs, so loads   */
/* stay in flight across the matrix ops (partial loadcnt waits).      */
/* OOB rows are address-clamped only: A row m feeds only D row m and  */
/* those rows are discarded at the guarded store.                     */
/* Requires Nout % 64 == 0, K % 4 == 0.                               */
/* ------------------------------------------------------------------ */
__global__ void ne_gemm_wmma(const float* __restrict__ A, int M, int K,
                             const float* __restrict__ W,
                             const float* __restrict__ bias,
                             float* __restrict__ C, int Nout, int relu)
{
  const int lane = threadIdx.x;
  const int half = lane >> 4, l16 = lane & 15;
  const int row  = blockIdx.x*16 + l16;
  const int rowc = (row < M) ? row : (M - 1);          /* clamp, not mask */
  const float* __restrict__ Arow = A + (size_t)rowc * K;
  const float* __restrict__ W0 = W + (size_t)(blockIdx.y*64 +  0 + l16) * K;
  const float* __restrict__ W1 = W + (size_t)(blockIdx.y*64 + 16 + l16) * K;
  const float* __restrict__ W2 = W + (size_t)(blockIdx.y*64 + 32 + l16) * K;
  const float* __restrict__ W3 = W + (size_t)(blockIdx.y*64 + 48 + l16) * K;

  v8f acc0 = {}, acc1 = {}, acc2 = {}, acc3 = {};
  const int kh = 2*half;
  /* prologue: fragments for k0 = 0 */
  v2f a  = *(const v2f*)(Arow + kh);
  v2f b0 = *(const v2f*)(W0 + kh);
  v2f b1 = *(const v2f*)(W1 + kh);
  v2f b2 = *(const v2f*)(W2 + kh);
  v2f b3 = *(const v2f*)(W3 + kh);

  for (int k0 = 0; k0 < K; k0 += 4) {
    /* issue next step's loads before this step's WMMAs (wraps to 0 at
       the tail: in-bounds, result unused) */
    const int kn = (k0 + 4 < K) ? (k0 + 4 + kh) : kh;
    v2f an  = *(const v2f*)(Arow + kn);
    v2f b0n = *(const v2f*)(W0 + kn);
    v2f b1n = *(const v2f*)(W1 + kn);
    v2f b2n = *(const v2f*)(W2 + kn);
    v2f b3n = *(const v2f*)(W3 + kn);

    acc0 = __builtin_amdgcn_wmma_f32_16x16x4_f32(false, a, false, b0, (short)0, acc0, false, false);
    acc1 = __builtin_amdgcn_wmma_f32_16x16x4_f32(false, a, false, b1, (short)0, acc1, false, false);
    acc2 = __builtin_amdgcn_wmma_f32_16x16x4_f32(false, a, false, b2, (short)0, acc2, false, false);
    acc3 = __builtin_amdgcn_wmma_f32_16x16x4_f32(false, a, false, b3, (short)0, acc3, false, false);

    a = an; b0 = b0n; b1 = b1n; b2 = b2n; b3 = b3n;
  }

  v8f accs[4] = {acc0, acc1, acc2, acc3};
  #pragma unroll
  for (int j = 0; j < 4; ++j) {
    const int ocol = blockIdx.y*64 + j*16 + l16;
    const float bv = bias ? bias[ocol] : 0.f;
    #pragma unroll
    for (int r = 0; r < 8; ++r) {
      int orow = blockIdx.x*16 + r + 8*half;
      if (orow < M) {
        float v = accs[j][r] + bv;
        if (relu) v = fmaxf(v, 0.f);
        C[(size_t)orow*Nout + ocol] = v;
      }
    }
  }
}

/* ------------------------------------------------------------------ */
/* Fused big GEMM: h0[50000,128] x conv1_w[1024,128]^T + b -> relu    */
/* then channel sums (BN stats) + per-(b,channel) max/min atomics.    */
/* Same double-buffered pipeline; M tile-exact (50000).               */
/* ------------------------------------------------------------------ */
__global__ void ne_gemm2_fused(const float* __restrict__ A,
                               const float* __restrict__ W,
                               const float* __restrict__ bias,
                               float* __restrict__ sum, float* __restrict__ sumsq,
                               unsigned* __restrict__ maxb, unsigned* __restrict__ minb)
{
  const int lane = threadIdx.x;
  const int half = lane >> 4, l16 = lane & 15;
  const int row  = blockIdx.x*16 + l16;
  const float* __restrict__ Arow = A + (size_t)row * C1;
  const float* __restrict__ W0 = W + (size_t)(blockIdx.y*64 +  0 + l16) * C1;
  const float* __restrict__ W1 = W + (size_t)(blockIdx.y*64 + 16 + l16) * C1;
  const float* __restrict__ W2 = W + (size_t)(blockIdx.y*64 + 32 + l16) * C1;
  const float* __restrict__ W3 = W + (size_t)(blockIdx.y*64 + 48 + l16) * C1;

  v8f acc0 = {}, acc1 = {}, acc2 = {}, acc3 = {};
  const int kh = 2*half;
  v2f a  = *(const v2f*)(Arow + kh);
  v2f b0 = *(const v2f*)(W0 + kh);
  v2f b1 = *(const v2f*)(W1 + kh);
  v2f b2 = *(const v2f*)(W2 + kh);
  v2f b3 = *(const v2f*)(W3 + kh);

  for (int k0 = 0; k0 < C1; k0 += 4) {
    const int kn = (k0 + 4 < C1) ? (k0 + 4 + kh) : kh;
    v2f an  = *(const v2f*)(Arow + kn);
    v2f b0n = *(const v2f*)(W0 + kn);
    v2f b1n = *(const v2f*)(W1 + kn);
    v2f b2n = *(const v2f*)(W2 + kn);
    v2f b3n = *(const v2f*)(W3 + kn);

    acc0 = __builtin_amdgcn_wmma_f32_16x16x4_f32(false, a, false, b0, (short)0, acc0, false, false);
    acc1 = __builtin_amdgcn_wmma_f32_16x16x4_f32(false, a, false, b1, (short)0, acc1, false, false);
    acc2 = __builtin_amdgcn_wmma_f32_16x16x4_f32(false, a, false, b2, (short)0, acc2, false, false);
    acc3 = __builtin_amdgcn_wmma_f32_16x16x4_f32(false, a, false, b3, (short)0, acc3, false, false);

    a = an; b0 = b0n; b1 = b1n; b2 = b2n; b3 = b3n;
  }

  v8f accs[4] = {acc0, acc1, acc2, acc3};
  #pragma unroll
  for (int j = 0; j < 4; ++j) {
    const int ocol = blockIdx.y*64 + j*16 + l16;
    const float bv = bias[ocol];
    float s = 0.f, ss = 0.f;
    #pragma unroll
    for (int r = 0; r < 8; ++r) {
      int orow = blockIdx.x*16 + r + 8*half;
      float v  = fmaxf(accs[j][r] + bv, 0.f);   /* relu >= 0: uint-bit atomics valid */
      s += v; ss += v*v;
      int b = orow / KNN;
      atomicMax(&maxb[(size_t)b*C2 + ocol], __float_as_uint(v));
      atomicMin(&minb[(size_t)b*C2 + ocol], __float_as_uint(v));
    }
    atomicAdd(&sum[ocol],   s);
    atomicAdd(&sumsq[ocol], ss);
  }
}

/* BN1 + max-over-K: feat[b,o] = (sel(max/min by sign) - m)*g/rstd + beta */
__global__ void ne_bn1_feat(const float* __restrict__ sum, const float* __restrict__ sumsq,
                            const unsigned* __restrict__ maxb, const unsigned* __restrict__ minb,
                            const float* __restrict__ g, const float* __restrict__ beta,
                            float* __restrict__ feat)
{
  int idx = blockIdx.x*blockDim.x + threadIdx.x;
  if (idx >= PBS*C2) return;
  int o = idx % C2;
  const float inv = 1.f / (float)ROWS;
  float m   = sum[o] * inv;
  float var = sumsq[o] * inv - m*m;
  float sc  = g[o] / sqrtf(var + EPSBN);
  float x   = (sc >= 0.f) ? __uint_as_float(maxb[idx]) : __uint_as_float(minb[idx]);
  feat[idx] = (x - m) * sc + beta[o];
}

/* BN over batch (500 rows), biased variance, in-place; 1 thread/channel */
__global__ void ne_bn_rows(float* __restrict__ X, int Cc,
                           const float* __restrict__ g, const float* __restrict__ beta)
{
  int o = blockIdx.x*blockDim.x + threadIdx.x;
  if (o >= Cc) return;
  float s = 0.f, ss = 0.f;
  for (int i = 0; i < PBS; ++i) { float v = X[(size_t)i*Cc + o]; s += v; ss += v*v; }
  float m   = s / (float)PBS;
  float var = ss / (float)PBS - m*m;
  float sc  = g[o] / sqrtf(var + EPSBN);
  float bo  = beta[o];
  for (int i = 0; i < PBS; ++i)
    X[(size_t)i*Cc + o] = (X[(size_t)i*Cc + o] - m) * sc + bo;
}

/* fc3 (256->3) + normal gather -> d_out [feat(1500) | normal[index](1500)] */
__global__ void ne_fc3_out(const float* __restrict__ X, const float* __restrict__ W,
                           const float* __restrict__ bias,
                           const float* __restrict__ normal, const int* __restrict__ index,
                           float* __restrict__ out)
{
  int b = blockIdx.x*blockDim.x + threadIdx.x;
  if (b >= PBS) return;
  #pragma unroll
  for (int j = 0; j < 3; ++j) {
    float acc = bias[j];
    for (int k = 0; k < 256; ++k) acc += X[(size_t)b*256 + k] * W[j*256 + k];
    out[b*3 + j] = acc;
  }
  int gi = index[b];
  out[PBS*3 + b*3 + 0] = normal[gi*3 + 0];
  out[PBS*3 + b*3 + 1] = normal[gi*3 + 1];
  out[PBS*3 + b*3 + 2] = normal[gi*3 + 2];
}

/* ------------------------------------------------------------------ */
extern "C" void kernel_launch(void* const* d_in, const int* in_sizes, int n_in,
                              void* d_out, int out_size, void* d_ws, size_t ws_size,
                              hipStream_t stream)
{
  (void)in_sizes; (void)n_in; (void)out_size; (void)ws_size;
  const float* pts     = (const float*)d_in[0];
  const float* normal  = (const float*)d_in[1];
  const float* kpp     = (const float*)d_in[2];
  const float* kpw     = (const float*)d_in[3];
  const float* sigma   = (const float*)d_in[4];
  const float* conv1_w = (const float*)d_in[5];
  const float* conv1_b = (const float*)d_in[6];
  const float* bn1_g   = (const float*)d_in[7];
  const float* bn1_b   = (const float*)d_in[8];
  const float* fc1_w   = (const float*)d_in[9];
  const float* fc1_b   = (const float*)d_in[10];
  const float* bn4_g   = (const float*)d_in[11];
  const float* bn4_b   = (const float*)d_in[12];
  const float* fc2_w   = (const float*)d_in[13];
  const float* fc2_b   = (const float*)d_in[14];
  const float* bn5_g   = (const float*)d_in[15];
  const float* bn5_b   = (const float*)d_in[16];
  const float* fc3_w   = (const float*)d_in[17];
  const float* fc3_b   = (const float*)d_in[18];
  const int*   index   = (const int*)d_in[19];

  char* ws = (char*)d_ws;
  size_t off = 0;
  auto carve = [&](size_t bytes) -> void* {
    void* p = ws + off;
    off = (off + bytes + 255) & ~(size_t)255;
    return p;
  };
  int*      nbr   = (int*)     carve((size_t)ROWS*4);
  float*    wf    = (float*)   carve((size_t)ROWS*KPAD*4);
  float*    wkT   = (float*)   carve((size_t)C1*KPAD*4);
  float*    h0    = (float*)   carve((size_t)ROWS*C1*4);
  float*    sum   = (float*)   carve((size_t)C2*4);
  float*    sumsq = (float*)   carve((size_t)C2*4);
  unsigned* maxb  = (unsigned*)carve((size_t)PBS*C2*4);
  unsigned* minb  = (unsigned*)carve((size_t)PBS*C2*4);
  float*    feat1 = (float*)   carve((size_t)PBS*C2*4);
  float*    t1    = (float*)   carve((size_t)PBS*512*4);
  float*    t2    = (float*)   carve((size_t)PBS*256*4);

  ne_init_kernel <<<dim3((PBS*C2+255)/256), dim3(256), 0, stream>>>(sum, sumsq, maxb, minb);
  ne_padwk_kernel<<<dim3((C1*KPAD+255)/256), dim3(256), 0, stream>>>(kpw, wkT);
  ne_knn_kernel  <<<dim3(PBS), dim3(256), 0, stream>>>(pts, index, nbr);
  ne_wf_kernel   <<<dim3((ROWS+255)/256), dim3(256), 0, stream>>>(pts, index, nbr, kpp, sigma, wf);

  /* KPConv linear: [50000,48] x [48,128] -> relu -> h0 */
  ne_gemm_wmma   <<<dim3(ROWS/16, C1/64), dim3(32), 0, stream>>>(wf, ROWS, KPAD, wkT, nullptr, h0, C1, 1);
  /* conv1 128->1024 fused relu + BN stats + max/min over K */
  ne_gemm2_fused <<<dim3(ROWS/16, C2/64), dim3(32), 0, stream>>>(h0, conv1_w, conv1_b, sum, sumsq, maxb, minb);
  ne_bn1_feat    <<<dim3((PBS*C2+255)/256), dim3(256), 0, stream>>>(sum, sumsq, maxb, minb, bn1_g, bn1_b, feat1);

  /* fc1 1024->512 + relu, BN over batch */
  ne_gemm_wmma   <<<dim3((PBS+15)/16, 512/64), dim3(32), 0, stream>>>(feat1, PBS, C2, fc1_w, fc1_b, t1, 512, 1);
  ne_bn_rows     <<<dim3(2), dim3(256), 0, stream>>>(t1, 512, bn4_g, bn4_b);
  /* fc2 512->256 + relu, BN over batch */
  ne_gemm_wmma   <<<dim3((PBS+15)/16, 256/64), dim3(32), 0, stream>>>(t1, PBS, 512, fc2_w, fc2_b, t2, 256, 1);
  ne_bn_rows     <<<dim3(1), dim3(256), 0, stream>>>(t2, 256, bn5_g, bn5_b);
  /* fc3 + normal gather */
  ne_fc3_out     <<<dim3(2), dim3(256), 0, stream>>>(t2, fc3_w, fc3_b, normal, index, (float*)d_out);
}